// HashEmbedder_80290118631529
// MI455X (gfx1250) — compile-verified
//
#include <hip/hip_runtime.h>
#include <stdint.h>

// ---------------------------------------------------------------------------
// Multiresolution hash-grid embedding (Instant-NGP style), CDNA5 / gfx1250.
//
// Layout: blockDim = 256 (8 wave32s). Thread t in a block handles
//   point  p = blockIdx.x*16 + (t >> 4)
//   level  l = t & 15
// so the block's 256 float2 results tile a contiguous 2KB span of `out`
// (out[b, l*2 + f], level-major within point) -> perfectly coalesced
// non-temporal b64 stores. Tables (64MB total) stay resident in the 192MB L2;
// the 128MB output is streamed past it with TH=NT.
//
// Coordinates for the block's 16 points (48 dwords) are staged into LDS with
// gfx1250 async load-to-LDS (ASYNCcnt) and then broadcast-read by the 16
// lanes sharing each point.
//
// All per-level constants (res, grid, 1/grid) are compile-time-folded
// __constant__ tables: no runtime f32 division anywhere in the kernel.
// ---------------------------------------------------------------------------

#define LOG2_T     19
#define HASH_MASK  ((1u << LOG2_T) - 1u)
#define NLV        16
#define PTS_PER_BLK 16
#define BLK        256

typedef __attribute__((ext_vector_type(2))) float f32x2;

// res[l]  = floor(16 * 2^(l/3))   (B_GROWTH = exp(ln(512/16)/15) = 2^(1/3))
// grid[l] = 2.8 / res[l]
// inv[l]  = res[l] / 2.8          (reciprocal, constant-folded)
#define RES_LIST(F) \
    F(16.f) F(20.f) F(25.f) F(32.f) F(40.f) F(50.f) F(64.f) F(80.f) \
    F(101.f) F(128.f) F(161.f) F(203.f) F(256.f) F(322.f) F(406.f) F(512.f)

#define GRID_OF(R) (2.8f / (R)),
#define INV_OF(R)  ((R) / 2.8f),

__constant__ float c_grid[NLV] = { RES_LIST(GRID_OF) };
__constant__ float c_inv[NLV]  = { RES_LIST(INV_OF) };

__device__ __forceinline__ f32x2 lerp2(f32x2 a, f32x2 b, float w) {
    f32x2 r;
    r.x = a.x * (1.0f - w) + b.x * w;
    r.y = a.y * (1.0f - w) + b.y * w;
    return r;
}

__global__ __launch_bounds__(BLK) void hashgrid_embed_kernel(
    const float* __restrict__ x,        // [B,3]
    const float* __restrict__ tables,   // [16, 2^19, 2]
    float* __restrict__ out,            // [B, 32]
    int B)
{
    __shared__ float sx[PTS_PER_BLK * 3];

    const int tid   = threadIdx.x;
    const int level = tid & (NLV - 1);
    const int pl    = tid >> 4;                       // point-in-block 0..15
    const long p    = (long)blockIdx.x * PTS_PER_BLK + pl;

    // ---- async-stage this block's coordinates into LDS (gfx1250 path) ----
    {
        const long gidx = (long)blockIdx.x * (PTS_PER_BLK * 3) + tid;
        if (tid < PTS_PER_BLK * 3 && gidx < (long)B * 3) {
            unsigned lds_off = (unsigned)(uintptr_t)(&sx[tid]); // low 32b = LDS offset
            uint64_t gaddr   = (uint64_t)(uintptr_t)(x + gidx);
            asm volatile("global_load_async_to_lds_b32 %0, %1, off"
                         :: "v"(lds_off), "v"(gaddr)
                         : "memory");
        }
    }
    asm volatile("s_wait_asynccnt 0" ::: "memory");   // ASYNCcnt -> 0 for issuing waves
    __syncthreads();                                  // make LDS visible block-wide

    if (p >= B) return;

    const float fx = sx[pl * 3 + 0];
    const float fy = sx[pl * 3 + 1];
    const float fz = sx[pl * 3 + 2];

    // ---- cell + trilinear weights (BOX = [-1.4, 1.4]) ----
    const float grid = c_grid[level];
    const float inv  = c_inv[level];

    const int bx = (int)floorf((fx + 1.4f) * inv);
    const int by = (int)floorf((fy + 1.4f) * inv);
    const int bz = (int)floorf((fz + 1.4f) * inv);

    const float wx = (fx - ((float)bx * grid - 1.4f)) * inv;
    const float wy = (fy - ((float)by * grid - 1.4f)) * inv;
    const float wz = (fz - ((float)bz * grid - 1.4f)) * inv;

    // ---- spatial hash: h = (cx*1) ^ (cy*2654435761) ^ (cz*805459861) ----
    // (c+1)*P == c*P + P  (mod 2^32)  -> 3 muls total for all 8 corners.
    const uint32_t x0 = (uint32_t)bx;
    const uint32_t x1 = x0 + 1u;
    const uint32_t y0 = (uint32_t)by * 2654435761u;
    const uint32_t y1 = y0 + 2654435761u;
    const uint32_t z0 = (uint32_t)bz * 805459861u;
    const uint32_t z1 = z0 + 805459861u;

    const f32x2* __restrict__ tb =
        (const f32x2*)tables + ((size_t)level << LOG2_T);

    // 8 independent b64 gathers (L2-resident tables) — issue all up front.
    const f32x2 e000 = tb[(x0 ^ y0 ^ z0) & HASH_MASK];
    const f32x2 e001 = tb[(x0 ^ y0 ^ z1) & HASH_MASK];
    const f32x2 e010 = tb[(x0 ^ y1 ^ z0) & HASH_MASK];
    const f32x2 e011 = tb[(x0 ^ y1 ^ z1) & HASH_MASK];
    const f32x2 e100 = tb[(x1 ^ y0 ^ z0) & HASH_MASK];
    const f32x2 e101 = tb[(x1 ^ y0 ^ z1) & HASH_MASK];
    const f32x2 e110 = tb[(x1 ^ y1 ^ z0) & HASH_MASK];
    const f32x2 e111 = tb[(x1 ^ y1 ^ z1) & HASH_MASK];

    // ---- trilinear blend: reduce x, then y, then z (matches reference) ----
    const f32x2 e00 = lerp2(e000, e100, wx);
    const f32x2 e01 = lerp2(e001, e101, wx);
    const f32x2 e10 = lerp2(e010, e110, wx);
    const f32x2 e11 = lerp2(e011, e111, wx);
    const f32x2 e0  = lerp2(e00, e10, wy);
    const f32x2 e1  = lerp2(e01, e11, wy);
    const f32x2 e   = lerp2(e0, e1, wz);

    // ---- streamed (non-temporal) coalesced store: out[p, level*2 .. +1] ----
    f32x2* dst = (f32x2*)(out + (size_t)p * (NLV * 2) + (size_t)level * 2);
    __builtin_nontemporal_store(e, dst);
}

extern "C" void kernel_launch(void* const* d_in, const int* in_sizes, int n_in,
                              void* d_out, int out_size, void* d_ws, size_t ws_size,
                              hipStream_t stream) {
    const float* x      = (const float*)d_in[0];   // [B,3] f32
    const float* tables = (const float*)d_in[1];   // [16, 2^19, 2] f32
    float*       out    = (float*)d_out;           // [B,32] f32

    const int B      = in_sizes[0] / 3;
    const int blocks = (B + PTS_PER_BLK - 1) / PTS_PER_BLK;

    hashgrid_embed_kernel<<<blocks, BLK, 0, stream>>>(x, tables, out, B);
}